// GNNBotnetDetector_70446053589555
// MI455X (gfx1250) — compile-verified
//
#include <hip/hip_runtime.h>
#include <hip/hip_bf16.h>
#include <math.h>

typedef __attribute__((ext_vector_type(2))) float v2f;
typedef __attribute__((ext_vector_type(8))) float v8f;

#define IN_DIM 128
#define HID 64

// ---------------------------------------------------------------------------
// Degree accumulation at destination (float atomics; exact for counts < 2^24)
// ---------------------------------------------------------------------------
__global__ void degree_kernel(const int* __restrict__ dst, float* __restrict__ deg, int E) {
    int e = blockIdx.x * blockDim.x + threadIdx.x;
    if (e < E) atomicAdd(&deg[dst[e]], 1.0f);
}

// dinv = rsqrt(deg + 1)   (+1 = self loop; always > 0)
__global__ void dinv_kernel(const float* __restrict__ deg, float* __restrict__ dinv, int N) {
    int i = blockIdx.x * blockDim.x + threadIdx.x;
    if (i < N) dinv[i] = rsqrtf(deg[i] + 1.0f);
}

// ---------------------------------------------------------------------------
// GEMM1: hw[N,64] = x[N,128] @ W1[128,64] via V_WMMA_F32_16X16X4_F32.
// Block = 128 threads = 4 waves; each wave owns one 16(M)x16(N) tile, so a
// block produces a full 16x64 slab. N (=100000) is a multiple of 16 -> no
// divergence, EXEC is all ones as WMMA requires.
//
// fp32 WMMA VGPR layouts (ISA 7.12.2):
//   A 16x4 : lane l -> row l%16 ; elem j -> K = 2*(l/16) + j
//   B 4x16 : lane l -> col l%16 ; elem j -> K = 2*(l/16) + j
//   D 16x16: VGPR t -> row t + 8*(l/16), col l%16
// ---------------------------------------------------------------------------
__global__ __launch_bounds__(128) void gemm1_wmma(const float* __restrict__ x,
                                                  const float* __restrict__ W1,
                                                  float* __restrict__ hw) {
    const int lane  = threadIdx.x & 31;
    const int wave  = threadIdx.x >> 5;       // 0..3 -> N tile
    const int row0  = blockIdx.x * 16;
    const int col0  = wave * 16;
    const int r     = lane & 15;
    const int khalf = (lane >> 4) * 2;        // 0 or 2

    const float* xrow = x + (size_t)(row0 + r) * IN_DIM;
    v8f c = {};

    #pragma unroll
    for (int kk = 0; kk < IN_DIM; kk += 4) {
        const int ka = kk + khalf;            // even -> 8B aligned
        v2f a = *(const v2f*)(xrow + ka);
        v2f b;
        b.x = W1[(size_t)ka * HID + col0 + r];
        b.y = W1[(size_t)(ka + 1) * HID + col0 + r];
        c = __builtin_amdgcn_wmma_f32_16x16x4_f32(
                /*neg_a=*/false, a, /*neg_b=*/false, b,
                /*c_mod=*/(short)0, c, /*reuse_a=*/false, /*reuse_b=*/false);
    }

    float* outp = hw + (size_t)(row0 + 8 * (lane >> 4)) * HID + col0 + r;
    #pragma unroll
    for (int t = 0; t < 8; ++t) outp[(size_t)t * HID] = c[t];
}

// Scalar tail for row counts not divisible by 16 (not hit for N=100000).
__global__ void gemm1_tail(const float* __restrict__ x, const float* __restrict__ W1,
                           float* __restrict__ hw, int rowStart, int N) {
    int idx = blockIdx.x * blockDim.x + threadIdx.x;
    int row = rowStart + idx / HID;
    int col = idx % HID;
    if (row >= N) return;
    float acc = 0.f;
    for (int k = 0; k < IN_DIM; ++k) acc += x[(size_t)row * IN_DIM + k] * W1[(size_t)k * HID + col];
    hw[(size_t)row * HID + col] = acc;
}

// ---------------------------------------------------------------------------
// Edge aggregation, layer 1: one wave32 per edge, each lane a float2 chunk of
// the 64-dim message. Gather from hw (L2-resident) + atomic scatter into agg.
// ---------------------------------------------------------------------------
__global__ void aggregate1_kernel(const int* __restrict__ src, const int* __restrict__ dst,
                                  const float* __restrict__ dinv, const float* __restrict__ hw,
                                  float* __restrict__ agg, int E) {
    size_t gt = (size_t)blockIdx.x * blockDim.x + threadIdx.x;
    int e    = (int)(gt >> 5);
    int lane = (int)(gt & 31);
    if (e >= E) return;
    int s = src[e], d = dst[e];
    float w = dinv[s] * dinv[d];
    float2 hv = *(const float2*)(hw + (size_t)s * HID + lane * 2);
    atomicAdd(&agg[(size_t)d * HID + lane * 2],     hv.x * w);
    atomicAdd(&agg[(size_t)d * HID + lane * 2 + 1], hv.y * w);
}

// h = relu(agg + self_loop + b1), written in place over hw
__global__ void finalize1_kernel(float* __restrict__ hw, const float* __restrict__ agg,
                                 const float* __restrict__ dinv, const float* __restrict__ b1,
                                 size_t total) {
    size_t i = (size_t)blockIdx.x * blockDim.x + threadIdx.x;
    if (i >= total) return;
    int node = (int)(i >> 6);
    int j    = (int)(i & 63);
    float di = dinv[node];
    float v  = agg[i] + hw[i] * di * di + b1[j];
    hw[i] = v > 0.f ? v : 0.f;
}

// ---------------------------------------------------------------------------
// GEMV: z[i] = h[i,:] . W2   (wave per node, shuffle reduction)
// ---------------------------------------------------------------------------
__global__ void gemm2_kernel(const float* __restrict__ h, const float* __restrict__ W2,
                             float* __restrict__ z, int N) {
    size_t gt = (size_t)blockIdx.x * blockDim.x + threadIdx.x;
    int node = (int)(gt >> 5);
    int lane = (int)(gt & 31);
    if (node >= N) return;
    float2 hv = *(const float2*)(h  + (size_t)node * HID + lane * 2);
    float2 wv = *(const float2*)(W2 + lane * 2);
    float p = hv.x * wv.x + hv.y * wv.y;
    #pragma unroll
    for (int o = 16; o > 0; o >>= 1) p += __shfl_xor(p, o, 32);
    if (lane == 0) z[node] = p;
}

// Edge aggregation, layer 2: one thread per edge, scalar atomic
__global__ void aggregate2_kernel(const int* __restrict__ src, const int* __restrict__ dst,
                                  const float* __restrict__ dinv, const float* __restrict__ z,
                                  float* __restrict__ outacc, int E) {
    int e = blockIdx.x * blockDim.x + threadIdx.x;
    if (e >= E) return;
    int s = src[e], d = dst[e];
    atomicAdd(&outacc[d], z[s] * dinv[s] * dinv[d]);
}

// out = sigmoid(outacc + self_loop + b2)
__global__ void finalize2_kernel(const float* __restrict__ outacc, const float* __restrict__ z,
                                 const float* __restrict__ dinv, const float* __restrict__ b2,
                                 float* __restrict__ out, int N) {
    int i = blockIdx.x * blockDim.x + threadIdx.x;
    if (i >= N) return;
    float di = dinv[i];
    float v = outacc[i] + z[i] * di * di + b2[0];
    out[i] = 1.0f / (1.0f + expf(-v));
}

// ---------------------------------------------------------------------------
extern "C" void kernel_launch(void* const* d_in, const int* in_sizes, int n_in,
                              void* d_out, int out_size, void* d_ws, size_t ws_size,
                              hipStream_t stream) {
    const float* x   = (const float*)d_in[0];
    const int*   ei  = (const int*)d_in[1];
    const float* W1  = (const float*)d_in[2];
    const float* b1  = (const float*)d_in[3];
    const float* W2  = (const float*)d_in[4];
    const float* b2  = (const float*)d_in[5];
    float* out = (float*)d_out;

    const int N = in_sizes[0] / IN_DIM;       // 100000
    const int E = in_sizes[1] / 2;            // 3200000
    const int* src = ei;
    const int* dst = ei + E;

    // Workspace layout (floats): deg[N] dinv[N] hw[N*64] agg[N*64] z[N] outacc[N]
    float* deg    = (float*)d_ws;
    float* dinv   = deg  + N;
    float* hw     = dinv + N;
    float* agg    = hw   + (size_t)N * HID;
    float* z      = agg  + (size_t)N * HID;
    float* outacc = z    + N;

    // Zero the accumulators every call (graph-capture safe memset nodes)
    hipMemsetAsync(deg,    0, (size_t)N * sizeof(float), stream);
    hipMemsetAsync(agg,    0, (size_t)N * HID * sizeof(float), stream);
    hipMemsetAsync(outacc, 0, (size_t)N * sizeof(float), stream);

    // Degrees -> dinv
    degree_kernel<<<(E + 255) / 256, 256, 0, stream>>>(dst, deg, E);
    dinv_kernel<<<(N + 255) / 256, 256, 0, stream>>>(deg, dinv, N);

    // GEMM1 via fp32 WMMA (full 16-row tiles) + scalar tail if ever needed
    int mainRows = (N / 16) * 16;
    if (mainRows > 0)
        gemm1_wmma<<<mainRows / 16, 128, 0, stream>>>(x, W1, hw);
    int tailRows = N - mainRows;
    if (tailRows > 0)
        gemm1_tail<<<(tailRows * HID + 255) / 256, 256, 0, stream>>>(x, W1, hw, mainRows, N);

    // Layer-1 aggregation: wave per edge
    {
        size_t total = (size_t)E * 32;
        aggregate1_kernel<<<(unsigned)((total + 255) / 256), 256, 0, stream>>>(src, dst, dinv, hw, agg, E);
    }
    finalize1_kernel<<<(unsigned)(((size_t)N * HID + 255) / 256), 256, 0, stream>>>(hw, agg, dinv, b1, (size_t)N * HID);

    // Layer 2
    {
        size_t total = (size_t)N * 32;
        gemm2_kernel<<<(unsigned)((total + 255) / 256), 256, 0, stream>>>(hw, W2, z, N);
    }
    aggregate2_kernel<<<(E + 255) / 256, 256, 0, stream>>>(src, dst, dinv, z, outacc, E);
    finalize2_kernel<<<(N + 255) / 256, 256, 0, stream>>>(outacc, z, dinv, b2, out, N);
}